// ManifoldRegularizationLoss_81003083202618
// MI455X (gfx1250) — compile-verified
//
#include <hip/hip_runtime.h>
#include <hip/hip_bf16.h>
#include <math.h>

typedef __attribute__((ext_vector_type(2))) float v2f;
typedef __attribute__((ext_vector_type(8))) float v8f;

#define NROWS   8192
#define DIM     256      // X feature dim (K of the Gram GEMM)
#define FDIM    512      // embedding dim
#define KNN     10
#define BAND    16       // rows per workgroup in k_knn
#define CHUNK   128      // columns per LDS chunk (8 waves * 16)
#define NCHUNKS (NROWS / CHUNK)
#define NSCAN   8        // scanner threads per row (8*16 = 128 scanners)
#define INF_F   3.0e38f
#define EPSL    1e-10f

// ---------------- workspace layout (float units) ----------------
#define OFF_SQN   0
#define OFF_KNND  8192
#define OFF_KNNI  90112
#define OFF_WTS   172032
#define OFF_DEG   253952
#define OFF_SIG   262144

// ---------------------------------------------------------------
__global__ void mrl_init(float* __restrict__ deg, float* __restrict__ sig,
                         float* __restrict__ out) {
  int t = blockIdx.x * blockDim.x + threadIdx.x;
  if (t < NROWS) deg[t] = 0.0f;
  if (t == 0) { sig[0] = 0.0f; sig[1] = 0.0f; out[0] = 0.0f; }
}

// one block (256 threads) per row; DIM == 256
__global__ void mrl_sqnorm(const float* __restrict__ X, float* __restrict__ sqn) {
  __shared__ float red[256];
  int row = blockIdx.x;
  float v = X[(size_t)row * DIM + threadIdx.x];
  red[threadIdx.x] = v * v;
  __syncthreads();
  for (int s = 128; s > 0; s >>= 1) {
    if (threadIdx.x < s) red[threadIdx.x] += red[threadIdx.x + s];
    __syncthreads();
  }
  if (threadIdx.x == 0) sqn[row] = red[0];
}

// ---------------------------------------------------------------
// Fused Gram (WMMA f32 16x16x4) + squared-distance + per-row top-10.
// Block = 256 threads (8 waves), one 16-row band per block.
// Selection is parallelized over 8 scanner threads per row; partial
// top-10s are merged once at the end (80 candidates per row).
__global__ __launch_bounds__(256)
void mrl_knn(const float* __restrict__ X, const float* __restrict__ sqn,
             float* __restrict__ knn_d, int* __restrict__ knn_i,
             float* __restrict__ sig) {
  __shared__ float Apanel[BAND * DIM];        // 16 KB: band rows of X
  __shared__ float chunk[BAND * CHUNK];       // 8 KB: d2 tile chunk (reused as cand_d)
  __shared__ int   candI[BAND * NSCAN * KNN]; // 5 KB: candidate indices
  __shared__ float sqnA[BAND];

  const int tid  = threadIdx.x;
  const int band = blockIdx.x;

  // Stage the 16-row A panel into LDS (1024 float4, 4 per thread).
  {
    const float4* Xv = (const float4*)(X + (size_t)band * BAND * DIM);
    float4*       Av = (float4*)Apanel;
    #pragma unroll
    for (int t = tid; t < (BAND * DIM) / 4; t += 256) Av[t] = Xv[t];
  }
  if (tid < BAND) sqnA[tid] = sqn[band * BAND + tid];
  __syncthreads();

  const int wave  = tid >> 5;
  const int lane  = tid & 31;
  const int mrow  = lane & 15;       // row-within-tile for A/B fragments, N for C
  const int khalf = (lane >> 4) * 2; // K sub-offset per ISA A/B 16x4 layout

  // scanner identity (threads 0..127): row srow, sub-scanner sp
  const int srow = tid >> 3;         // 0..31 (only srow<16 used)
  const int sp   = tid & 7;          // 0..7

  // per-scanner top-10 state
  float bd[KNN];
  int   bi[KNN];
  #pragma unroll
  for (int s = 0; s < KNN; ++s) { bd[s] = INF_F; bi[s] = -1; }

  const float* Arow = Apanel + mrow * DIM + khalf;

  for (int cc = 0; cc < NCHUNKS; ++cc) {
    const int j0 = cc * CHUNK + wave * BAND;

    // prefetch next chunk's B rows (global_prefetch_b8)
    if (lane < BAND && (j0 + CHUNK) < NROWS)
      __builtin_prefetch((const void*)(X + (size_t)(j0 + CHUNK + lane) * DIM), 0, 0);

    const float* Brow = X + (size_t)(j0 + mrow) * DIM + khalf;

    // Gram tile: two interleaved accumulators to break the D->C chain.
    v8f c0 = {}; v8f c1 = {};
    #pragma unroll 4
    for (int k0 = 0; k0 < DIM; k0 += 8) {
      v2f a0 = *(const v2f*)(Arow + k0);
      v2f b0 = *(const v2f*)(Brow + k0);
      c0 = __builtin_amdgcn_wmma_f32_16x16x4_f32(false, a0, false, b0,
                                                 (short)0, c0, false, false);
      v2f a1 = *(const v2f*)(Arow + k0 + 4);
      v2f b1 = *(const v2f*)(Brow + k0 + 4);
      c1 = __builtin_amdgcn_wmma_f32_16x16x4_f32(false, a1, false, b1,
                                                 (short)0, c1, false, false);
    }

    // d2 = |xi|^2 + |xj|^2 - 2*gram ; C layout: lane<16 -> M=v, else M=v+8; N=mrow
    const float snB = sqn[j0 + mrow];
    const int   gj  = j0 + mrow;
    #pragma unroll
    for (int v = 0; v < 8; ++v) {
      int M  = (lane < 16) ? v : (v + 8);
      int gi = band * BAND + M;
      float g  = c0[v] + c1[v];
      float d2 = sqnA[M] + snB - 2.0f * g;
      d2 = (gi == gj) ? INF_F : fmaxf(d2, 0.0f);
      chunk[M * CHUNK + wave * BAND + mrow] = d2;
    }
    __syncthreads();

    // 128 scanner threads: each handles 16 strided columns of its row
    if (tid < BAND * NSCAN) {
      const int jbase = cc * CHUNK;
      #pragma unroll
      for (int t = 0; t < CHUNK / NSCAN; ++t) {
        const int c2 = sp + NSCAN * t;
        float v = chunk[srow * CHUNK + c2];
        if (v < bd[KNN - 1]) {
          int jj = jbase + c2;
          int pos = KNN - 1;
          #pragma unroll
          for (int s = KNN - 2; s >= 0; --s) if (v < bd[s]) pos = s;
          #pragma unroll
          for (int s = KNN - 1; s > 0; --s)
            if (s > pos) { bd[s] = bd[s - 1]; bi[s] = bi[s - 1]; }
          bd[pos] = v; bi[pos] = jj;
        }
      }
    }
    __syncthreads();
  }

  // dump per-scanner candidates (reuse `chunk` as cand_d: 1280 <= 2048 floats)
  if (tid < BAND * NSCAN) {
    #pragma unroll
    for (int s = 0; s < KNN; ++s) {
      chunk[tid * KNN + s] = bd[s];
      candI[tid * KNN + s] = bi[s];
    }
  }
  __syncthreads();

  // final merge: thread r folds its row's 80 candidates into a top-10
  if (tid < BAND) {
    float fd[KNN]; int fi[KNN];
    #pragma unroll
    for (int s = 0; s < KNN; ++s) { fd[s] = INF_F; fi[s] = -1; }
    for (int p = 0; p < NSCAN; ++p) {
      #pragma unroll
      for (int s = 0; s < KNN; ++s) {
        float v = chunk[(tid * NSCAN + p) * KNN + s];
        if (v < fd[KNN - 1]) {
          int jj = candI[(tid * NSCAN + p) * KNN + s];
          int pos = KNN - 1;
          #pragma unroll
          for (int q = KNN - 2; q >= 0; --q) if (v < fd[q]) pos = q;
          #pragma unroll
          for (int q = KNN - 1; q > 0; --q)
            if (q > pos) { fd[q] = fd[q - 1]; fi[q] = fi[q - 1]; }
          fd[pos] = v; fi[pos] = jj;
        }
      }
    }
    int gi = band * BAND + tid;
    float ssum = 0.0f;
    #pragma unroll
    for (int s = 0; s < KNN; ++s) {
      float d = sqrtf(fmaxf(fd[s], 0.0f));
      knn_d[gi * KNN + s] = d;
      knn_i[gi * KNN + s] = fi[s];
      ssum += d;
    }
    atomicAdd(&sig[0], ssum);
  }
}

// ---------------------------------------------------------------
__global__ void mrl_sigma(float* __restrict__ sig) {
  float s = sig[0] / (float)(NROWS * KNN);
  sig[1] = 1.0f / (2.0f * s * s);
}

// weights + deduplicated degree (reproduces max-symmetrization)
__global__ void mrl_wdeg(const float* __restrict__ knn_d, const int* __restrict__ knn_i,
                         float* __restrict__ wts, float* __restrict__ deg,
                         const float* __restrict__ sig) {
  int i = blockIdx.x * blockDim.x + threadIdx.x;
  if (i >= NROWS) return;
  const float inv2s2 = sig[1];
  #pragma unroll
  for (int s = 0; s < KNN; ++s) {
    float d = knn_d[i * KNN + s];
    int   j = knn_i[i * KNN + s];
    float w = __expf(-d * d * inv2s2);
    wts[i * KNN + s] = w;
    bool mutual = false;
    #pragma unroll
    for (int t = 0; t < KNN; ++t) mutual = mutual || (knn_i[j * KNN + t] == i);
    if (!(mutual && (i > j))) {   // count each undirected edge exactly once
      atomicAdd(&deg[i], w);
      atomicAdd(&deg[j], w);
    }
  }
}

// diagonal part: one wave32 per row, loss += (deg/(deg+eps)) * ||f_i||^2
__global__ void mrl_loss_diag(const float* __restrict__ emb,
                              const float* __restrict__ deg,
                              float* __restrict__ out) {
  int wv = threadIdx.x >> 5, lane = threadIdx.x & 31;
  int i = blockIdx.x * 8 + wv;
  const float* f = emb + (size_t)i * FDIM;
  float s = 0.0f;
  for (int c = lane; c < FDIM; c += 32) { float v = f[c]; s += v * v; }
  #pragma unroll
  for (int m = 16; m > 0; m >>= 1) s += __shfl_xor(s, m, 32);
  if (lane == 0) {
    float dg = deg[i];
    atomicAdd(out, (dg / (dg + EPSL)) * s);
  }
}

// off-diagonal: one wave32 per directed edge; dedup to undirected, factor 2
__global__ void mrl_loss_edge(const float* __restrict__ emb,
                              const int* __restrict__ knn_i,
                              const float* __restrict__ wts,
                              const float* __restrict__ deg,
                              float* __restrict__ out) {
  int e = blockIdx.x * 8 + (threadIdx.x >> 5);
  if (e >= NROWS * KNN) return;
  int lane = threadIdx.x & 31;
  int i = e / KNN;
  int j = knn_i[e];
  bool mutual = false;
  #pragma unroll
  for (int t = 0; t < KNN; ++t) mutual = mutual || (knn_i[j * KNN + t] == i);
  if (mutual && (i > j)) return;   // undirected dedup (wave-uniform)
  float w = wts[e];
  float scale = -2.0f * w * rsqrtf(deg[i] + EPSL) * rsqrtf(deg[j] + EPSL);
  const float* fi = emb + (size_t)i * FDIM;
  const float* fj = emb + (size_t)j * FDIM;
  float s = 0.0f;
  for (int c = lane; c < FDIM; c += 32) s += fi[c] * fj[c];
  #pragma unroll
  for (int m = 16; m > 0; m >>= 1) s += __shfl_xor(s, m, 32);
  if (lane == 0) atomicAdd(out, scale * s);
}

// ---------------------------------------------------------------
extern "C" void kernel_launch(void* const* d_in, const int* in_sizes, int n_in,
                              void* d_out, int out_size, void* d_ws, size_t ws_size,
                              hipStream_t stream) {
  const float* emb = (const float*)d_in[0];   // [8192, 512]
  const float* X   = (const float*)d_in[1];   // [8192, 256]
  float* out = (float*)d_out;

  float* wsf   = (float*)d_ws;
  float* sqn   = wsf + OFF_SQN;
  float* knn_d = wsf + OFF_KNND;
  int*   knn_i = (int*)(wsf + OFF_KNNI);
  float* wts   = wsf + OFF_WTS;
  float* deg   = wsf + OFF_DEG;
  float* sig   = wsf + OFF_SIG;

  mrl_init<<<NROWS / 256, 256, 0, stream>>>(deg, sig, out);
  mrl_sqnorm<<<NROWS, 256, 0, stream>>>(X, sqn);
  mrl_knn<<<NROWS / BAND, 256, 0, stream>>>(X, sqn, knn_d, knn_i, sig);
  mrl_sigma<<<1, 1, 0, stream>>>(sig);
  mrl_wdeg<<<NROWS / 256, 256, 0, stream>>>(knn_d, knn_i, wts, deg, sig);
  mrl_loss_diag<<<NROWS / 8, 256, 0, stream>>>(emb, deg, out);
  mrl_loss_edge<<<(NROWS * KNN) / 8, 256, 0, stream>>>(emb, knn_i, wts, deg, out);
}